// RearrangedFeaturesMap_35107062677847
// MI455X (gfx1250) — compile-verified
//
#include <hip/hip_runtime.h>

// Problem constants (from reference): B=16, N=300, C=512, FS=4, TH=TW=20
#define Bn   16
#define Nn   300
#define Cn   512
#define POSN 400   // TH*TW
#define THW  20    // TW
#define OH   80    // TH*FS

typedef float v4f __attribute__((ext_vector_type(4)));

// --- Kernel A1: initialize inverse map to "empty" -------------------------
__global__ __launch_bounds__(256) void rfm_init_inv(int* __restrict__ inv) {
    int i = blockIdx.x * blockDim.x + threadIdx.x;
    if (i < Bn * POSN) inv[i] = -1;
}

// --- Kernel A2: scatter token->grid inverse map: inv[b][pos] = n | d8<<20 -
__global__ __launch_bounds__(256) void rfm_scatter_inv(const int* __restrict__ t_pos,
                                                       const int* __restrict__ d8,
                                                       int* __restrict__ inv) {
    int i = blockIdx.x * blockDim.x + threadIdx.x;
    if (i >= Bn * Nn) return;
    int b   = i / Nn;
    int n   = i - b * Nn;
    int pos = t_pos[i];                 // unique per batch (permutation)
    inv[b * POSN + pos] = n | (d8[i] << 20);
}

// --- Kernel B: output-driven gather; one thread = one 16B output segment --
// out[b][c][4*th+fy][4*tw+fx]; seg order = (b, c, y, tw) so out addr = seg*16B.
__global__ __launch_bounds__(256) void rfm_gather(const float* __restrict__ features,
                                                  const float* __restrict__ backend,
                                                  const int*   __restrict__ inv,
                                                  float*       __restrict__ out) {
    int seg = blockIdx.x * blockDim.x + threadIdx.x;   // exact grid: no bounds check
    int tw  = seg % THW;
    int t1  = seg / THW;
    int y   = t1 % OH;
    int t2  = t1 / OH;
    int c   = t2 % Cn;
    int b   = t2 / Cn;
    int fy  = y & 3;
    int th  = y >> 2;

    int code = inv[b * POSN + th * THW + tw];
    v4f v;
    if (code < 0) {
        // unfilled grid cell -> backend_feature[c] broadcast
        float bf = backend[c];
        v = (v4f){bf, bf, bf, bf};
    } else {
        int n = code & 0xFFFF;
        int d = code >> 20;
        // 64B source tile: features[b][n][c][0..15]
        const float* tile = features + ((((size_t)b * Nn + n) * Cn + c) << 4);
        // dihedral: fmaps[d][fy*4+fx] = (d&1) ? bcol*4+a : a*4+bcol
        int a = (d & 4) ? (3 - fy) : fy;
        if ((d & 1) == 0) {
            // row-contiguous source: single 128-bit load (+ optional reverse)
            v4f t = *(const v4f*)(tile + (a << 2));
            v = (d & 2) ? (v4f){t.w, t.z, t.y, t.x} : t;
        } else {
            // transposed source: stride-16B column gather
            float e0 = tile[a], e1 = tile[4 + a], e2 = tile[8 + a], e3 = tile[12 + a];
            v = (d & 2) ? (v4f){e3, e2, e1, e0} : (v4f){e0, e1, e2, e3};
        }
    }
    // 210MB write-once stream: non-temporal 128-bit store (keep L2 for gathers)
    __builtin_nontemporal_store(v, (v4f*)out + seg);
}

extern "C" void kernel_launch(void* const* d_in, const int* in_sizes, int n_in,
                              void* d_out, int out_size, void* d_ws, size_t ws_size,
                              hipStream_t stream) {
    const float* features = (const float*)d_in[0];   // (B,N,C,4,4) f32
    const int*   t_pos    = (const int*)  d_in[1];   // (B,N) i32
    const int*   d8       = (const int*)  d_in[2];   // (B,N) i32
    const float* backend  = (const float*)d_in[3];   // (1,C,1,1) f32

    int* inv = (int*)d_ws;                           // B*400 ints = 25.6 KB

    rfm_init_inv   <<<(Bn * POSN + 255) / 256, 256, 0, stream>>>(inv);
    rfm_scatter_inv<<<(Bn * Nn   + 255) / 256, 256, 0, stream>>>(t_pos, d8, inv);

    const int total_segs = Bn * Cn * OH * THW;       // 13,107,200 (exact /256)
    rfm_gather<<<total_segs / 256, 256, 0, stream>>>(features, backend, inv,
                                                     (float*)d_out);
}